// SpectralPooling_62483184222926
// MI455X (gfx1250) — compile-verified
//
#include <hip/hip_runtime.h>
#include <math.h>

typedef __attribute__((ext_vector_type(2))) float v2f;
typedef __attribute__((ext_vector_type(8))) float v8f;

// ---------------------------------------------------------------------------
// Kernel 0: build combined DCT->truncate->IDCT matrix A (32 x 64)
//   A[r][w] = sum_{p<24} dct32[p][r] * dct64[p][w]
// ---------------------------------------------------------------------------
__global__ __launch_bounds__(256) void build_A_kernel(float* __restrict__ Aout) {
    int idx = blockIdx.x * 256 + threadIdx.x;      // 0 .. 2047
    if (idx >= 32 * 64) return;
    int r = idx >> 6;                              // output spatial index (0..31)
    int w = idx & 63;                              // input spatial index  (0..63)
    const float PI = 3.14159265358979323846f;
    float acc = 0.0f;
    for (int p = 0; p < 24; ++p) {
        float s32 = (p == 0) ? sqrtf(1.0f / 32.0f) : sqrtf(2.0f / 32.0f);
        float s64 = (p == 0) ? sqrtf(1.0f / 64.0f) : sqrtf(2.0f / 64.0f);
        float c32 = cosf(PI * ((float)r + 0.5f) * (float)p / 32.0f);
        float c64 = cosf(PI * ((float)w + 0.5f) * (float)p / 64.0f);
        acc += c32 * s32 * c64 * s64;
    }
    Aout[idx] = acc;
}

// ---------------------------------------------------------------------------
// Generic batched GEMM: C(32 x N) = A(32 x 64) * B(64 x N) per batch item.
// One wave computes a full 32 x 16 stripe of C using v_wmma_f32_16x16x4_f32.
// MODE 0: standard row-major store  C[batch][m][n]
// MODE 1: final stage — batch=(b,p,q), N=32 (channels), m=r; store to
//         out[b][r][q][p][c] (the reference's (0,3,2,1,4) transpose).
// NT     : use non-temporal loads for B (streaming, read-once input).
// ---------------------------------------------------------------------------
template <int MODE, bool NT>
__global__ __launch_bounds__(256) void dct_gemm_kernel(
    const float* __restrict__ Aglob,   // 32x64
    const float* __restrict__ B,       // batch x 64 x N (rows contiguous)
    float* __restrict__ C,
    int N, int ntilesPerBatch, int totalTiles)
{
    __shared__ float A_lds[32 * 64];
    for (int i = threadIdx.x; i < 32 * 64; i += 256) A_lds[i] = Aglob[i];
    __syncthreads();

    const int lane = threadIdx.x & 31;
    const int wave = threadIdx.x >> 5;
    const int t = blockIdx.x * 8 + wave;
    if (t >= totalTiles) return;                 // uniform per-wave exit, EXEC stays all-1s

    const int batchIdx = t / ntilesPerBatch;
    const int ntile    = t - batchIdx * ntilesPerBatch;
    const int n0       = ntile * 16;

    const int row  = lane & 15;                  // M/N index within tile
    const int koff = (lane >> 4) << 1;           // 0 or 2 (K sub-offset)
    const int col  = n0 + row;

    // Hoist all A fragments (2 M-tiles x 16 k-steps) into registers:
    // lane<16: {A[m0+row][k0], A[m0+row][k0+1]}, lane>=16: k0+2, k0+3  (ds_load_b64)
    v2f afrag[2][16];
#pragma unroll
    for (int mt = 0; mt < 2; ++mt) {
#pragma unroll
        for (int kk = 0; kk < 16; ++kk) {
            const float* ap = &A_lds[(mt * 16 + row) * 64 + kk * 4 + koff];
            v2f a; a.x = ap[0]; a.y = ap[1];
            afrag[mt][kk] = a;
        }
    }

    const float* Bb = B + (size_t)batchIdx * 64 * (size_t)N;
    v8f acc0 = {}; v8f acc1 = {};

#pragma unroll
    for (int kk = 0; kk < 16; ++kk) {
        const int krow = kk * 4 + koff;
        const float* bp0 = Bb + (size_t)krow * N + col;
        const float* bp1 = bp0 + N;
        v2f b;
        if (NT) { b.x = __builtin_nontemporal_load(bp0);
                  b.y = __builtin_nontemporal_load(bp1); }
        else    { b.x = *bp0; b.y = *bp1; }
        acc0 = __builtin_amdgcn_wmma_f32_16x16x4_f32(false, afrag[0][kk], false, b,
                                                     (short)0, acc0, false, false);
        acc1 = __builtin_amdgcn_wmma_f32_16x16x4_f32(false, afrag[1][kk], false, b,
                                                     (short)0, acc1, false, false);
    }

    const int rbase = (lane >> 4) << 3;          // 0 or 8: D row = rbase + v (+ m0)
    if (MODE == 0) {
        float* Cb = C + (size_t)batchIdx * 32 * (size_t)N;
#pragma unroll
        for (int v = 0; v < 8; ++v) {
            Cb[(size_t)(rbase + v)      * N + col] = acc0[v];
            Cb[(size_t)(16 + rbase + v) * N + col] = acc1[v];
        }
    } else {
        // batchIdx = (b*32 + p)*32 + q ; m = r ; n = channel c
        const int b  = batchIdx >> 10;
        const int pq = batchIdx & 1023;
        const int p  = pq >> 5;
        const int q  = pq & 31;
#pragma unroll
        for (int v = 0; v < 8; ++v) {
            int r0 = rbase + v;
            size_t i0 = ((((size_t)b * 32 + r0)        * 32 + q) * 32 + p) * 32 + col;
            size_t i1 = ((((size_t)b * 32 + (16 + r0)) * 32 + q) * 32 + p) * 32 + col;
            C[i0] = acc0[v];
            C[i1] = acc1[v];
        }
    }
}

// ---------------------------------------------------------------------------
// Launch: build A, then 3 batched-GEMM passes (d, h, w) with transposed store.
//   x : (4, 64, 64, 64, 32) f32
//   z1: (4, 32, 64*64*32)   f32  = 67.1 MB  (fits in 192 MB L2)
//   z2: (4, 32, 32, 64*32)  f32  = 33.6 MB
//   out:(4, 32, 32, 32, 32) f32
// ---------------------------------------------------------------------------
extern "C" void kernel_launch(void* const* d_in, const int* in_sizes, int n_in,
                              void* d_out, int out_size, void* d_ws, size_t ws_size,
                              hipStream_t stream) {
    (void)in_sizes; (void)n_in; (void)out_size; (void)ws_size;
    const float* x  = (const float*)d_in[0];
    float* out      = (float*)d_out;

    char* ws = (char*)d_ws;
    float* A  = (float*)ws;                                   // 8 KB
    float* z1 = (float*)(ws + 8192);                          // 4*32*131072*4 = 64 MiB
    float* z2 = (float*)(ws + 8192 + (size_t)4*32*131072*4);  // 4*32*32*2048*4 = 32 MiB

    // Build the fused DCT/IDCT operator A (32x64).
    build_A_kernel<<<8, 256, 0, stream>>>(A);

    // Stage 1: contract d.  batch=4, N=64*64*32=131072, ntiles=8192 -> 32768 wave-tiles.
    dct_gemm_kernel<0, true><<<32768 / 8, 256, 0, stream>>>(
        A, x, z1, 131072, 8192, 32768);

    // Stage 2: contract h.  batch=4*32=128, N=64*32=2048, ntiles=128 -> 16384 wave-tiles.
    dct_gemm_kernel<0, false><<<16384 / 8, 256, 0, stream>>>(
        A, z1, z2, 2048, 128, 16384);

    // Stage 3: contract w + transposed store. batch=4*32*32=4096, N=32, ntiles=2 -> 8192.
    dct_gemm_kernel<1, false><<<8192 / 8, 256, 0, stream>>>(
        A, z2, out, 32, 2, 8192);
}